// WeightsOnlyAttention_12532714569732
// MI455X (gfx1250) — compile-verified
//
#include <hip/hip_runtime.h>

typedef __attribute__((ext_vector_type(2))) float v2f;
typedef __attribute__((ext_vector_type(8))) float v8f;

#define B_   4
#define P_   1024
#define E_   768
#define S_   16
#define H_   64
#define NCOL 2048            // 2*S*H
#define NEGC (-1.0e15f)

// -----------------------------------------------------------------------------
// Kernel 1: qk = x @ W + b  (4096x768 @ 768x2048), split into Q (scaled) and K.
// One wave32 per 32x16 output super-tile (two M-tiles share each B fragment),
// V_WMMA_F32_16X16X4_F32 over K=768.
// A (16x4 f32): lane = M row (lane&15), vgpr pair = K (kb = (lane>>4)*2).
// B (4x16 f32): lane = N col (lane&15), vgpr pair = K.
// C/D: vgpr v holds M = v + 8*(lane>>4), N = lane&15.
// -----------------------------------------------------------------------------
__global__ __launch_bounds__(256) void proj_kernel(const float* __restrict__ x,
                                                   const float* __restrict__ W,
                                                   const float* __restrict__ bias,
                                                   float* __restrict__ qbuf,
                                                   float* __restrict__ kbuf) {
  const int lane = threadIdx.x & 31;
  const int wv   = threadIdx.x >> 5;
  const int task = blockIdx.x * 8 + wv;       // 16384 tasks total
  const int mt2  = task >> 7;                 // 128 M super-tiles (32 rows)
  const int nt   = task & 127;                // 128 N tiles
  const int m0   = mt2 << 5;
  const int n0   = nt << 4;
  const int l15  = lane & 15;
  const int kb   = (lane >> 4) << 1;          // 0 or 2

  const float* arow0 = x + (size_t)(m0 + l15) * E_;
  const float* arow1 = arow0 + (size_t)16 * E_;
  const float* bcol  = W + n0 + l15;

  v8f acc0 = {};
  v8f acc1 = {};
  for (int e = 0; e < E_; e += 4) {
    v2f a0 = *(const v2f*)(arow0 + e + kb);
    v2f a1 = *(const v2f*)(arow1 + e + kb);
    v2f bv;
    bv.x = bcol[(size_t)(e + kb) * NCOL];
    bv.y = bcol[(size_t)(e + kb + 1) * NCOL];
    __builtin_prefetch(bcol + (size_t)(e + kb + 4) * NCOL, 0, 3);
    acc0 = __builtin_amdgcn_wmma_f32_16x16x4_f32(false, a0, false, bv,
                                                 (short)0, acc0, false, false);
    acc1 = __builtin_amdgcn_wmma_f32_16x16x4_f32(false, a1, false, bv,
                                                 (short)0, acc1, false, false);
  }

  const int n  = n0 + l15;
  const int c  = n >> 10;                     // 0 = Q, 1 = K
  const int sh = n & 1023;
  const int s  = sh >> 6;
  const int h  = sh & 63;
  const float bi    = bias[n];
  const float scale = (c == 0) ? 0.125f : 1.0f;   // 1/sqrt(H) for Q
  float* dst = (c == 0) ? qbuf : kbuf;
  const int mrow0 = m0 + ((lane >> 4) << 3);

  for (int v = 0; v < 8; ++v) {
    // first M-tile
    int m  = mrow0 + v;
    int bb = m >> 10;
    int p  = m & 1023;
    dst[(((size_t)(bb * S_ + s)) * P_ + p) * H_ + h] = (acc0[v] + bi) * scale;
    // second M-tile (+16 rows)
    m  += 16;
    bb  = m >> 10;
    p   = m & 1023;
    dst[(((size_t)(bb * S_ + s)) * P_ + p) * H_ + h] = (acc1[v] + bi) * scale;
  }
}

// -----------------------------------------------------------------------------
// Kernel 2: per (b, s, 16-row query tile): scores = Q Kt via WMMA into LDS,
// causal mask + softmax over k, coalesced write of w to d_out.
// 256 threads = 8 waves; wave w computes k-tiles {w, w+8, ...} (wave-uniform,
// EXEC all-1s around WMMA). Q fragments are preloaded once per wave into
// registers and reused across all k-tiles. Then each wave handles 2 softmax
// rows with wave32 shfl reductions.
// -----------------------------------------------------------------------------
__global__ __launch_bounds__(256) void attn_kernel(const float* __restrict__ qbuf,
                                                   const float* __restrict__ kbuf,
                                                   const float* __restrict__ mask,
                                                   float* __restrict__ out) {
  __shared__ float sc[16 * 1024];             // 64 KB: 16 q-rows x 1024 k
  const int lane = threadIdx.x & 31;
  const int wv   = threadIdx.x >> 5;
  const int qt = blockIdx.x & 63;
  const int s  = (blockIdx.x >> 6) & 15;
  const int b  = blockIdx.x >> 10;
  const int q0 = qt << 4;
  const int l15 = lane & 15;
  const int kb  = (lane >> 4) << 1;
  const int r0  = (lane >> 4) << 3;

  const size_t head = (size_t)(b * S_ + s) * P_;
  const float* Q = qbuf + (head + q0) * (size_t)H_;
  const float* K = kbuf + head * (size_t)H_;
  const float* Qrow = Q + (size_t)l15 * H_;

  // Preload A fragments (loop-invariant across k-tiles): 16 x v2f = 32 VGPRs.
  v2f qf[16];
#pragma unroll
  for (int i = 0; i < 16; ++i) qf[i] = *(const v2f*)(Qrow + (i << 2) + kb);

  for (int kt = wv; kt < 64; kt += 8) {
    const int k0 = kt << 4;
    if (k0 <= q0 + 15) {                      // causal: skip fully-masked tiles
      const float* Krow = K + (size_t)(k0 + l15) * H_;
      v8f acc = {};
#pragma unroll
      for (int i = 0; i < 16; ++i) {
        v2f bv = *(const v2f*)(Krow + (i << 2) + kb);
        acc = __builtin_amdgcn_wmma_f32_16x16x4_f32(false, qf[i], false, bv,
                                                    (short)0, acc, false, false);
      }
      for (int v = 0; v < 8; ++v)
        sc[(r0 + v) * 1024 + k0 + l15] = acc[v];
    } else {
      for (int v = 0; v < 8; ++v)
        sc[(r0 + v) * 1024 + k0 + l15] = 0.0f;
    }
  }
  __syncthreads();

  for (int rr = 0; rr < 2; ++rr) {
    const int r  = wv * 2 + rr;
    const int qg = q0 + r;
    float* srow = sc + r * 1024;
    const float* mrow = mask + (size_t)qg * P_;

    float mx = -3.0e38f;
    for (int k = lane; k < 1024; k += 32) {
      float v = srow[k] + (1.0f - mrow[k]) * NEGC;
      srow[k] = v;
      mx = fmaxf(mx, v);
    }
    for (int off = 16; off; off >>= 1) mx = fmaxf(mx, __shfl_xor(mx, off, 32));

    float sum = 0.0f;
    for (int k = lane; k < 1024; k += 32) {
      float e = __expf(srow[k] - mx);
      srow[k] = e;
      sum += e;
    }
    for (int off = 16; off; off >>= 1) sum += __shfl_xor(sum, off, 32);
    const float inv = 1.0f / sum;

    float* orow = out + ((head + qg) << 10);
    for (int k = lane; k < 1024; k += 32) orow[k] = srow[k] * inv;
  }
}

// -----------------------------------------------------------------------------
// Kernel 3: in-place layer-norm across the S (head) axis at every (b,q,k).
// One thread owns all 16 head values of one (b,q,k) -> race-free in-place.
// -----------------------------------------------------------------------------
__global__ __launch_bounds__(256) void ln_kernel(float* __restrict__ out,
                                                 const float* __restrict__ lns,
                                                 const float* __restrict__ lnb) {
  const int t = blockIdx.x * 256 + threadIdx.x;   // 4,194,304 threads
  const int k = t & 1023;
  const int q = (t >> 10) & 1023;
  const int b = t >> 20;
  const size_t base = (size_t)b * ((size_t)S_ * 1048576) + (size_t)q * 1024 + k;

  float v[16];
  float sum = 0.0f, sq = 0.0f;
  for (int s = 0; s < 16; ++s) {
    float xv = out[base + (size_t)s * 1048576];
    v[s] = xv;
    sum += xv;
    sq  += xv * xv;
  }
  const float mu   = sum * 0.0625f;
  const float var  = sq * 0.0625f - mu * mu;      // population variance (ddof=0)
  const float rstd = rsqrtf(var + 1e-5f);
  for (int s = 0; s < 16; ++s)
    out[base + (size_t)s * 1048576] = (v[s] - mu) * rstd * lns[s] + lnb[s];
}

// -----------------------------------------------------------------------------
extern "C" void kernel_launch(void* const* d_in, const int* in_sizes, int n_in,
                              void* d_out, int out_size, void* d_ws, size_t ws_size,
                              hipStream_t stream) {
  (void)in_sizes; (void)n_in; (void)out_size; (void)ws_size;
  const float* x    = (const float*)d_in[0];
  const float* mask = (const float*)d_in[1];
  const float* W    = (const float*)d_in[2];
  const float* bias = (const float*)d_in[3];
  const float* lns  = (const float*)d_in[4];
  const float* lnb  = (const float*)d_in[5];
  float* out  = (float*)d_out;
  float* qbuf = (float*)d_ws;                               // 16 MB
  float* kbuf = qbuf + (size_t)B_ * S_ * P_ * H_;           // +16 MB

  // 16384 32x16 super-tiles, 8 waves/block
  proj_kernel<<<2048, 256, 0, stream>>>(x, W, bias, qbuf, kbuf);
  // B * S * (P/16) = 4096 blocks
  attn_kernel<<<4096, 256, 0, stream>>>(qbuf, kbuf, mask, out);
  // B * P * P / 256 = 16384 blocks
  ln_kernel<<<16384, 256, 0, stream>>>(out, lns, lnb);
}